// VectorQuantizer_9440338116773
// MI455X (gfx1250) — compile-verified
//
#include <hip/hip_runtime.h>
#include <hip/hip_bf16.h>

// ---------------------------------------------------------------------------
// VQ-VAE vector quantizer, fused GEMM(+argmin)+gather+loss for gfx1250.
//   z:   [8,4096,256] f32   emb: [8192,256] f32
//   out: [ z_q (8388608 f32) | loss (1 f32) | idx (32768 f32) ]
// Core: f32 WMMA 16x16x4 (argmin is too precision-sensitive for bf16/fp8),
// A-stationary in VGPRs, B double-buffered into LDS via the Tensor Data
// Mover (TDM) with hardware row padding; TENSORcnt-synchronized.
// Four independent accumulator chains per chunk for WMMA/LDS latency cover.
// ---------------------------------------------------------------------------

typedef __attribute__((ext_vector_type(2))) float        v2f;
typedef __attribute__((ext_vector_type(4))) float        f4v;
typedef __attribute__((ext_vector_type(8))) float        v8f;
typedef __attribute__((ext_vector_type(4))) unsigned int v4u;
typedef __attribute__((ext_vector_type(4))) int          v4i;
typedef __attribute__((ext_vector_type(8))) int          v8i;

#define VQ_D        256
#define VQ_NE       8192
#define VQ_T        32768          // 8*4096 tokens
#define VQ_NB       64             // codes per LDS chunk (double buffered)
#define VQ_LDS_STR  260            // padded row stride (floats): conflict-free b64 reads
#define VQ_NCHUNK   (VQ_NE / VQ_NB)
#define VQ_BUF_FL   (VQ_NB * VQ_LDS_STR)
#define VQ_NUMEL    8388608.0f     // 8*4096*256
#define VQ_BETA     0.25f

#if defined(__has_builtin)
#if __has_builtin(__builtin_amdgcn_tensor_load_to_lds)
#define VQ_HAVE_TDM 1
#endif
#endif
#ifndef VQ_HAVE_TDM
#define VQ_HAVE_TDM 0
#endif

// ---------------------------------------------------------------------------
// Kernel 1: per-code squared norms  ||e||^2  (one wave per code row)
// ---------------------------------------------------------------------------
__global__ __launch_bounds__(256) void vq_norms_kernel(
    const float* __restrict__ emb, float* __restrict__ norms) {
  const int lane = threadIdx.x & 31;
  const int row  = blockIdx.x * 8 + (threadIdx.x >> 5);
  const float* er = emb + (size_t)row * VQ_D + lane * 8;
  float s = 0.0f;
#pragma unroll
  for (int j = 0; j < 8; ++j) { float v = er[j]; s += v * v; }
#pragma unroll
  for (int m = 1; m < 32; m <<= 1) s += __shfl_xor(s, m, 32);
  if (lane == 0) norms[row] = s;
}

// ---------------------------------------------------------------------------
// Chunk staging: TDM descriptor per wave (8 emb rows -> padded LDS tile),
// falling back to a synchronous cooperative copy if the builtin is absent.
// ---------------------------------------------------------------------------
__device__ __forceinline__ void vq_stage_chunk(const float* __restrict__ emb,
                                               int cbase, float* __restrict__ buf,
                                               unsigned int bufLdsByte, int tid) {
#if VQ_HAVE_TDM
  const int wave = tid >> 5;
  const unsigned long long ga =
      (unsigned long long)(uintptr_t)(emb + (size_t)(cbase + wave * 8) * VQ_D);
  const unsigned int ldsByte =
      bufLdsByte + (unsigned int)(wave * 8 * VQ_LDS_STR * 4);
  // Wave-uniform descriptor dwords -> SGPRs.
  unsigned int g0w1 = (unsigned int)__builtin_amdgcn_readfirstlane((int)ldsByte);
  unsigned int g0w2 = (unsigned int)__builtin_amdgcn_readfirstlane((int)(unsigned int)ga);
  unsigned int g0w3 =
      ((unsigned int)__builtin_amdgcn_readfirstlane((int)(unsigned int)(ga >> 32)) &
       0x01ffffffu) | (2u << 30);                       // addr[56:32] | type=2
  v4u g0 = {1u /*count=1*/, g0w1, g0w2, g0w3};
  // Group1: data_size=4B, pad_enable, pad_interval=256 DW, pad_amount=4 DW
  const unsigned int c1w0 = (2u << 16) | (1u << 20) | (7u << 22) | (3u << 25);
  const unsigned int c1w1 = (256u & 0xffffu) << 16;     // tensor_dim0 lo -> [63:48]
  const unsigned int c1w2 = (256u >> 16) | ((8192u & 0xffffu) << 16); // td0 hi | td1 lo
  const unsigned int c1w3 = (8192u >> 16) | (256u << 16);             // td1 hi | tile_dim0=256
  const unsigned int c1w4 = 8u;                         // tile_dim1=8 rows, tile_dim2=0
  const unsigned int c1w5 = 256u;                       // tensor_dim0_stride lo
  v8i g1 = {(int)c1w0, (int)c1w1, (int)c1w2, (int)c1w3,
            (int)c1w4, (int)c1w5, 0, 0};
  v4i z4 = {0, 0, 0, 0};
#if __clang_major__ >= 23
  v8i z8 = {0, 0, 0, 0, 0, 0, 0, 0};
  __builtin_amdgcn_tensor_load_to_lds(g0, g1, z4, z4, z8, 0);
#else
  __builtin_amdgcn_tensor_load_to_lds(g0, g1, z4, z4, 0);
#endif
  (void)buf;
#else
  const int row = tid >> 2;                   // 64 rows, 4 threads/row
  const int seg = (tid & 3) * 64;
  const float* src = emb + (size_t)(cbase + row) * VQ_D + seg;
  float* dst = buf + row * VQ_LDS_STR + seg;  // row base 16B aligned (1040B stride)
#pragma unroll
  for (int i = 0; i < 16; ++i) ((f4v*)dst)[i] = ((const f4v*)src)[i];
  (void)bufLdsByte;
#endif
}

__device__ __forceinline__ void vq_stage_wait() {
#if VQ_HAVE_TDM
#if defined(__has_builtin) && __has_builtin(__builtin_amdgcn_s_wait_tensorcnt)
  __builtin_amdgcn_s_wait_tensorcnt(0);
#else
  asm volatile("s_wait_tensorcnt 0x0" ::: "memory");
#endif
#endif
}

// ---------------------------------------------------------------------------
// Kernel 2: fused distance-GEMM + argmin + gather + loss partials.
// grid: 256 blocks x 256 threads. Block = 8 waves, wave = 16 tokens.
// ---------------------------------------------------------------------------
__global__ __launch_bounds__(256) void vq_main_kernel(
    const float* __restrict__ z, const float* __restrict__ emb,
    const float* __restrict__ norms, float* __restrict__ zq,
    float* __restrict__ idxOut, float* __restrict__ lossPartials) {
  extern __shared__ float lds[];              // 2 * VQ_BUF_FL floats (~130 KB)

  const int lane = threadIdx.x & 31;
  const int wave = threadIdx.x >> 5;
  const int tok0 = blockIdx.x * 128 + wave * 16;
  const int half = lane >> 4;                 // 0: K pair {0,1}, 1: K pair {2,3}
  const int mcol = lane & 15;                 // A row (token) / B col (code) id
  const unsigned int ldsBase = (unsigned int)(uintptr_t)lds; // low 32 bits = LDS offset

  // ---- A-stationary: this wave's 16 token rows in WMMA A layout ----------
  v2f areg[64];
  {
    const float* zr = z + (size_t)(tok0 + mcol) * VQ_D + 2 * half;
#pragma unroll
    for (int s = 0; s < 64; ++s) areg[s] = *(const v2f*)(zr + 4 * s);
  }

  float minScore[8];
  int   minIdx[8];
#pragma unroll
  for (int r = 0; r < 8; ++r) { minScore[r] = 3.0e38f; minIdx[r] = 0; }

  // ---- prologue: stage chunk 0 into buffer 0 -----------------------------
  vq_stage_chunk(emb, 0, lds, ldsBase, threadIdx.x);

  for (int c = 0; c < VQ_NCHUNK; ++c) {
    vq_stage_wait();                          // this wave's TDM slice landed
    __syncthreads();                          // all slices landed; prev reads done
    {
      // Unconditional: last iteration re-stages chunk 0 into the buffer that
      // is not being read; drained by the implicit wait-idle at s_endpgm.
      const int cn = (c + 1) & (VQ_NCHUNK - 1);
      const int nb = (c + 1) & 1;
      vq_stage_chunk(emb, cn * VQ_NB, lds + nb * VQ_BUF_FL,
                     ldsBase + (unsigned int)(nb * VQ_BUF_FL * 4), threadIdx.x);
    }
    const float* buf = lds + (c & 1) * VQ_BUF_FL;
    const int cbase = c * VQ_NB;

    // ---- whole 64-code chunk as one body: 4 independent acc chains ------
    const float* bp[4];
    v8f acc[4];
#pragma unroll
    for (int t = 0; t < 4; ++t) {
      bp[t]  = buf + (t * 16 + mcol) * VQ_LDS_STR + 2 * half;
      acc[t] = (v8f){};
    }
#pragma unroll
    for (int s = 0; s < 64; ++s) {
#pragma unroll
      for (int t = 0; t < 4; ++t) {
        v2f bb = *(const v2f*)(bp[t] + 4 * s);
        acc[t] = __builtin_amdgcn_wmma_f32_16x16x4_f32(
            false, areg[s], false, bb, (short)0, acc[t], false, false);
      }
    }
#pragma unroll
    for (int t = 0; t < 4; ++t) {
      const int n = cbase + t * 16 + mcol;
      const float nrm = norms[n];
#pragma unroll
      for (int r = 0; r < 8; ++r) {
        // score = ||e||^2 - 2 z.e   (||z||^2 per-token constant: irrelevant)
        float sc = nrm - 2.0f * acc[t][r];
        bool tk = sc < minScore[r];
        minScore[r] = tk ? sc : minScore[r];
        minIdx[r]   = tk ? n : minIdx[r];
      }
    }
  }

  // ---- min-reduce across the 16 lanes of each half (xor bits 0..3) -------
#pragma unroll
  for (int r = 0; r < 8; ++r) {
#pragma unroll
    for (int m = 1; m < 16; m <<= 1) {
      float os = __shfl_xor(minScore[r], m, 32);
      int   oi = __shfl_xor(minIdx[r], m, 32);
      bool take = (os < minScore[r]) || (os == minScore[r] && oi < minIdx[r]);
      minScore[r] = take ? os : minScore[r];
      minIdx[r]   = take ? oi : minIdx[r];
    }
  }
  // C layout: VGPR r holds token (r + 8*half)
  if (mcol == 0) {
#pragma unroll
    for (int r = 0; r < 8; ++r)
      idxOut[tok0 + half * 8 + r] = (float)minIdx[r];
  }

  // ---- gather z_q = emb[idx], accumulate loss partial --------------------
  float lacc = 0.0f;
#pragma unroll
  for (int m = 0; m < 16; ++m) {
    const int srcLane = (m < 8) ? 0 : 16;
    const int e = __shfl(minIdx[m & 7], srcLane, 32);
    const float* er = emb + (size_t)e * VQ_D + lane * 8;
    const float* zr = z + (size_t)(tok0 + m) * VQ_D + lane * 8;
    float* qr = zq + (size_t)(tok0 + m) * VQ_D + lane * 8;
#pragma unroll
    for (int j = 0; j < 8; ++j) {
      float ev = er[j];
      float d  = ev - zr[j];
      lacc += d * d;
      qr[j] = ev;
    }
  }
#pragma unroll
  for (int m = 1; m < 32; m <<= 1) lacc += __shfl_xor(lacc, m, 32);
  if (lane == 0) lossPartials[blockIdx.x * 8 + wave] = lacc;
}

// ---------------------------------------------------------------------------
// Kernel 3: deterministic fixed-order reduction of 2048 loss partials.
// ---------------------------------------------------------------------------
__global__ __launch_bounds__(256) void vq_loss_kernel(
    const float* __restrict__ partials, float* __restrict__ lossOut) {
  __shared__ float red[256];
  float s = 0.0f;
#pragma unroll
  for (int i = 0; i < 8; ++i) s += partials[threadIdx.x + 256 * i];
  red[threadIdx.x] = s;
  __syncthreads();
  for (int w = 128; w > 0; w >>= 1) {
    if ((int)threadIdx.x < w) red[threadIdx.x] += red[threadIdx.x + w];
    __syncthreads();
  }
  if (threadIdx.x == 0)
    lossOut[0] = red[0] * ((1.0f + VQ_BETA) / VQ_NUMEL);
}

// ---------------------------------------------------------------------------
extern "C" void kernel_launch(void* const* d_in, const int* in_sizes, int n_in,
                              void* d_out, int out_size, void* d_ws, size_t ws_size,
                              hipStream_t stream) {
  const float* z   = (const float*)d_in[0];   // 8*4096*256
  const float* emb = (const float*)d_in[1];   // 8192*256

  float* out    = (float*)d_out;
  float* zq     = out;                        // [0, 8388608)
  float* loss   = out + 8388608;              // [8388608]
  float* idxOut = out + 8388609;              // [8388609, +32768)

  float* norms    = (float*)d_ws;             // 8192 floats
  float* partials = norms + VQ_NE;            // 2048 floats

  // 1) code norms (one wave per code row)
  vq_norms_kernel<<<VQ_NE / 8, 256, 0, stream>>>(emb, norms);

  // 2) fused distance GEMM + argmin + gather + loss partials
  const size_t ldsBytes = (size_t)2 * VQ_BUF_FL * sizeof(float); // ~130 KB
  vq_main_kernel<<<VQ_T / 128, 256, ldsBytes, stream>>>(
      z, emb, norms, zq, idxOut, partials);

  // 3) deterministic loss finish
  vq_loss_kernel<<<1, 256, 0, stream>>>(partials, loss);
}